// WeightedSenseEmbedding_35021163332165
// MI455X (gfx1250) — compile-verified
//
#include <hip/hip_runtime.h>
#include <math.h>

typedef float v2f __attribute__((ext_vector_type(2)));
typedef float v4f __attribute__((ext_vector_type(4)));
typedef float v8f __attribute__((ext_vector_type(8)));

#define SZ    8192      // B*L
#define DIM   300
#define KS    5
#define CTXN  10
#define NNEG  5
#define ROWS  15        // CTXN + NNEG
#define PITCH 308       // A-tile pitch (dwords): float4-aligned, bank-conflict-free
#define EPSF  1e-20f

__global__ __launch_bounds__(256) void wse_main(
    const int* __restrict__ pivots,
    const int* __restrict__ contexts,
    const int* __restrict__ neg,
    const float* __restrict__ gumbel,
    const float* __restrict__ senseW,
    const float* __restrict__ ctxW,
    float* __restrict__ emb_out,
    float* __restrict__ loss_part)
{
  __shared__ __align__(16) float pivLDS[DIM * KS];   // 1500 f
  __shared__ __align__(16) float AT[16 * PITCH];     // 16 x 308 f
  __shared__ __align__(16) float BT[(DIM / 2) * 32]; // 150 pairs x 16 n x 2 par
  __shared__ float dots[16 * 16];
  __shared__ float att[KS];
  __shared__ float terms[ROWS * KS];

  const int s = blockIdx.x;
  const int t = threadIdx.x;

  // ---- stage pivot sense row (1500 floats) into LDS, float4 coalesced ----
  {
    const v4f* src = (const v4f*)(senseW + (size_t)pivots[s] * (DIM * KS));
    v4f* dst = (v4f*)pivLDS;
    for (int i = t; i < (DIM * KS) / 4; i += 256) dst[i] = src[i];  // 375
  }
  // ---- stage 10 ctx rows + 5 neg rows + 1 zero row into A-tile ----
  for (int i = t; i < 16 * (DIM / 4); i += 256) {    // 1200 float4s
    int r = i / (DIM / 4);
    int q = i - r * (DIM / 4);
    v4f v = {0.f, 0.f, 0.f, 0.f};
    if (r < ROWS) {
      int w = (r < CTXN) ? contexts[s * CTXN + r] : neg[s * NNEG + (r - CTXN)];
      v = ((const v4f*)(ctxW + (size_t)w * DIM))[q];
    }
    *(v4f*)&AT[r * PITCH + q * 4] = v;
  }
  __syncthreads();

  // ---- build interleaved, zero-padded B tile: BT[p*32 + n*2 + par] =
  //      piv[d = 2p+par][n] for n<5, else 0. Unconditional clamped load. ----
  for (int i = t; i < (DIM / 2) * 32; i += 256) {    // 4800
    int p   = i >> 5;
    int r   = i & 31;
    int n   = r >> 1;
    int par = r & 1;
    int d   = 2 * p + par;
    int nc  = (n < KS) ? n : 0;                      // clamp: load stays unconditional
    float v = pivLDS[d * KS + nc];
    BT[i] = (n < KS) ? v : 0.f;
  }
  __syncthreads();

  // ---- wave 0: padded 16x16 GEMM over K=300 via V_WMMA_F32_16X16X4_F32 ----
  if (t < 32) {
    const int lane = t;
    const int m    = lane & 15;          // A row / B column / C column
    const int hi   = lane >> 4;
    const int koff = hi * 2;             // hi half-wave holds K+2..K+3
    const float* Arow = &AT[m * PITCH + koff];
    const float* Brow = &BT[hi * 32 + m * 2];
    v8f acc = {0.f, 0.f, 0.f, 0.f, 0.f, 0.f, 0.f, 0.f};
#pragma unroll
    for (int k0 = 0; k0 < DIM; k0 += 4) {
      v2f a = *(const v2f*)(Arow + k0);              // K = k0+koff, +1
      v2f b = *(const v2f*)(Brow + (k0 / 2) * 32);   // B[kk][n], B[kk+1][n]
      acc = __builtin_amdgcn_wmma_f32_16x16x4_f32(false, a, false, b,
                                                  (short)0, acc, false, false);
    }
    const int mbase = hi * 8;                        // C: VGPR j -> M = j + 8*hi
#pragma unroll
    for (int j = 0; j < 8; ++j) dots[(mbase + j) * 16 + m] = acc[j];
  }
  __syncthreads();

  // ---- gumbel softmax attention (5 logits, one thread) ----
  if (t == 0) {
    float y[KS];
    float mx = -1e30f;
    for (int k = 0; k < KS; ++k) {
      float prod = 0.f;
      for (int c = 0; c < CTXN; ++c) prod += dots[c * 16 + k];
      prod *= (1.0f / CTXN);                         // ctx.mean reuses ctx dots
      float g  = gumbel[s * KS + k];
      float yy = prod - 0.5f * logf(-logf(g + EPSF) + EPSF);  // SCALE=0.5
      y[k] = yy * 2.0f;                              // divide by TAU=0.5
      mx = fmaxf(mx, y[k]);
    }
    float sum = 0.f;
    for (int k = 0; k < KS; ++k) { float e = expf(y[k] - mx); att[k] = e; sum += e; }
    float inv = 1.f / sum;
    for (int k = 0; k < KS; ++k) att[k] *= inv;
  }
  __syncthreads();

  // ---- emb[s, d] = sum_k piv[d][k] * att[k] ----
  for (int d = t; d < DIM; d += 256) {
    float a0 = 0.f;
#pragma unroll
    for (int k = 0; k < KS; ++k) a0 += pivLDS[d * KS + k] * att[k];
    emb_out[(size_t)s * DIM + d] = a0;
  }
  // ---- 75 log-sigmoid loss terms ----
  if (t < ROWS * KS) {
    int r = t / KS, k = t - r * KS;
    float dot = dots[r * 16 + k];
    float x  = (r < CTXN) ? dot : -dot;              // negs use sigmoid(-dot)
    float sg = 1.f / (1.f + expf(-x));
    terms[t] = att[k] * logf(EPSF + sg);
  }
  __syncthreads();
  if (t == 0) {                                      // deterministic serial sum
    float sum = 0.f;
    for (int i = 0; i < ROWS * KS; ++i) sum += terms[i];
    loss_part[s] = sum;
  }
}

__global__ __launch_bounds__(256) void wse_reduce(const float* __restrict__ part,
                                                  float* __restrict__ loss_out)
{
  __shared__ float sm[256];
  int t = threadIdx.x;
  float acc = 0.f;
  for (int i = t; i < SZ; i += 256) acc += part[i];   // fixed order per thread
  sm[t] = acc;
  __syncthreads();
  for (int off = 128; off > 0; off >>= 1) {           // fixed-shape tree
    if (t < off) sm[t] += sm[t + off];
    __syncthreads();
  }
  if (t == 0) loss_out[0] = -sm[0];                   // loss = -(sum corr + sum negs)
}

extern "C" void kernel_launch(void* const* d_in, const int* in_sizes, int n_in,
                              void* d_out, int out_size, void* d_ws, size_t ws_size,
                              hipStream_t stream)
{
  const int*   pivots   = (const int*)d_in[0];
  const int*   contexts = (const int*)d_in[1];
  const int*   neg      = (const int*)d_in[2];
  const float* gumbel   = (const float*)d_in[3];
  const float* senseW   = (const float*)d_in[4];
  const float* ctxW     = (const float*)d_in[5];
  float* out = (float*)d_out;
  float* ws  = (float*)d_ws;   // 8192 floats of partial losses

  wse_main<<<SZ, 256, 0, stream>>>(pivots, contexts, neg, gumbel, senseW, ctxW,
                                   out, ws);
  wse_reduce<<<1, 256, 0, stream>>>(ws, out + (size_t)SZ * DIM);
}